// Ridge_Transformer_56925496541572
// MI455X (gfx1250) — compile-verified
//
#include <hip/hip_runtime.h>
#include <hip/hip_bf16.h>

// Problem constants (match the reference).
#define BB   2
#define NN   1024
#define DD   384
#define HH   8
#define DFFC 512
#define LLC  4
#define HDC  48
#define RBFC 16
#define FCK  387           // D + 3
#define FCK_PAD 388        // padded to multiple of 4

typedef float v2f __attribute__((ext_vector_type(2)));
typedef float v8f __attribute__((ext_vector_type(8)));

__device__ __forceinline__ v8f wmma_f32(v2f a, v2f b, v8f c) {
    // D = A(16x4,f32) * B(4x16,f32) + C(16x16,f32)
    return __builtin_amdgcn_wmma_f32_16x16x4_f32(
        /*neg_a=*/false, a, /*neg_b=*/false, b,
        /*c_mod=*/(short)0, c, /*reuse_a=*/false, /*reuse_b=*/false);
}

// ---------------------------------------------------------------------------
// Pack weight [Ksrc, Ncols] -> pair-interleaved [Kpad/2, Ncols, 2], zero pad.
// Lane pair {k, k+1} at a column becomes one aligned 64-bit load.
// ---------------------------------------------------------------------------
__global__ void pack_pairs_kernel(const float* __restrict__ in,
                                  float* __restrict__ out,
                                  int Ksrc, int Kpad, int Ncols) {
    int idx = blockIdx.x * blockDim.x + threadIdx.x;
    int total = Kpad * Ncols;
    if (idx >= total) return;
    int k = idx / Ncols, c = idx - k * Ncols;
    float val = (k < Ksrc) ? in[(size_t)k * Ncols + c] : 0.0f;
    out[((size_t)(k >> 1) * Ncols + c) * 2 + (k & 1)] = val;
}

// ---------------------------------------------------------------------------
// LayerNorm: one block (128 threads) per row of 384.
// ---------------------------------------------------------------------------
__global__ void ln_kernel(const float* __restrict__ x,
                          const float* __restrict__ g,
                          const float* __restrict__ b,
                          float* __restrict__ out) {
    int row = blockIdx.x;
    int t = threadIdx.x;
    const float* xr = x + (size_t)row * DD;
    float v0 = xr[t], v1 = xr[t + 128], v2 = xr[t + 256];
    __shared__ float rs[128], rq[128];
    rs[t] = v0 + v1 + v2;
    rq[t] = v0 * v0 + v1 * v1 + v2 * v2;
    __syncthreads();
    for (int off = 64; off > 0; off >>= 1) {
        if (t < off) { rs[t] += rs[t + off]; rq[t] += rq[t + off]; }
        __syncthreads();
    }
    float mean = rs[0] * (1.0f / DD);
    float var  = rq[0] * (1.0f / DD) - mean * mean;
    float inv  = rsqrtf(var + 1e-5f);
    float* o = out + (size_t)row * DD;
    o[t]       = (v0 - mean) * inv * g[t]       + b[t];
    o[t + 128] = (v1 - mean) * inv * g[t + 128] + b[t + 128];
    o[t + 256] = (v2 - mean) * inv * g[t + 256] + b[t + 256];
}

// ---------------------------------------------------------------------------
// fp32 WMMA GEMM, 16x64 strip per wave (4 accumulators, A reused 4x).
// A[M,K] row-major; Wp = pair-interleaved packed weight [K/2][N][2].
// K must be a multiple of 4. flags: 1=relu, 2=add residual, 4=packed-pair out.
// ---------------------------------------------------------------------------
__global__ void gemm4_kernel(const float* __restrict__ A, int lda,
                             const float* __restrict__ Wp, int ldw,
                             const float* __restrict__ bias,
                             const float* __restrict__ addend,
                             float* __restrict__ Out, int ldo,
                             int K, int flags) {
    int tn = blockIdx.x, tm = blockIdx.y;
    int lane = threadIdx.x;
    int hi = lane >> 4, lo = lane & 15;
    int row = tm * 16 + lo;
    int colbase = tn * 64 + lo;
    const float* arow = A + (size_t)row * lda;
    v8f acc[4] = {{}, {}, {}, {}};
    for (int k0 = 0; k0 < K; k0 += 4) {
        int ka = k0 + 2 * hi;
        v2f a = *(const v2f*)(arow + ka);
        const float* wp = Wp + ((size_t)((k0 >> 1) + hi) * ldw + colbase) * 2;
        v2f b0 = *(const v2f*)(wp);
        v2f b1 = *(const v2f*)(wp + 32);
        v2f b2 = *(const v2f*)(wp + 64);
        v2f b3 = *(const v2f*)(wp + 96);
        acc[0] = wmma_f32(a, b0, acc[0]);
        acc[1] = wmma_f32(a, b1, acc[1]);
        acc[2] = wmma_f32(a, b2, acc[2]);
        acc[3] = wmma_f32(a, b3, acc[3]);
    }
#pragma unroll
    for (int t4 = 0; t4 < 4; ++t4) {
        int col = tn * 64 + 16 * t4 + lo;
        float bv = bias[col];
#pragma unroll
        for (int r = 0; r < 8; ++r) {
            int orow = tm * 16 + r + 8 * hi;
            float val = acc[t4][r] + bv;
            if (flags & 1) val = fmaxf(val, 0.0f);
            if (flags & 2) val += addend[(size_t)orow * ldo + col];
            if (flags & 4)
                Out[((size_t)(orow >> 1) * ldo + col) * 2 + (orow & 1)] = val;
            else
                Out[(size_t)orow * ldo + col] = val;
        }
    }
}

// ---------------------------------------------------------------------------
// RBF distance bias: dis[b,i,j] = sum_r exp(-((d-mu_r)/sigma)^2)*Wmlp[r]+bmlp
// ---------------------------------------------------------------------------
__global__ void dis_kernel(const float* __restrict__ distance,
                           const float* __restrict__ wmlp,
                           const float* __restrict__ bmlp,
                           float* __restrict__ dis) {
    size_t idx = (size_t)blockIdx.x * blockDim.x + threadIdx.x;
    if (idx >= (size_t)BB * NN * NN) return;
    float d = distance[idx];
    const float inv_sigma = 16.0f / 20.0f;     // 1/1.25
    float acc = bmlp[0];
#pragma unroll
    for (int r = 0; r < RBFC; ++r) {
        float mu = (float)r * (20.0f / 15.0f); // linspace(0,20,16)
        float t = (d - mu) * inv_sigma;
        acc += __expf(-t * t) * wmlp[r];
    }
    dis[idx] = acc;
}

// ---------------------------------------------------------------------------
// Scores: S[b,h,i,j] = (q_i . k_j)/sqrt(48) + dis[b,i,j]; mask -> -1e9.
// One wave per 16(i) x 64(j) strip; both operands contiguous along HD -> b64.
// ---------------------------------------------------------------------------
__global__ void scores_kernel(const float* __restrict__ q,
                              const float* __restrict__ k,
                              const float* __restrict__ dis,
                              const int* __restrict__ mask,
                              float* __restrict__ scores) {
    int tn = blockIdx.x, tm = blockIdx.y;
    int bh = blockIdx.z;
    int b = bh / HH, h = bh % HH;
    int lane = threadIdx.x;
    int hi = lane >> 4, lo = lane & 15;
    const float* qb = q + (size_t)b * NN * DD + h * HDC;
    const float* kb = k + (size_t)b * NN * DD + h * HDC;
    const float* qr = qb + (size_t)(tm * 16 + lo) * DD;
    int coljbase = tn * 64 + lo;
    v8f acc[4] = {{}, {}, {}, {}};
#pragma unroll
    for (int k0 = 0; k0 < HDC; k0 += 4) {
        int ka = k0 + 2 * hi;
        v2f a = *(const v2f*)(qr + ka);
#pragma unroll
        for (int t4 = 0; t4 < 4; ++t4) {
            v2f bm = *(const v2f*)(kb + (size_t)(coljbase + 16 * t4) * DD + ka);
            acc[t4] = wmma_f32(a, bm, acc[t4]);
        }
    }
    const float inv_scale = 0.14433756729740643f;   // 1/sqrt(48)
    size_t sbase = (size_t)(b * HH + h) * NN;
#pragma unroll
    for (int t4 = 0; t4 < 4; ++t4) {
        int colj = tn * 64 + 16 * t4 + lo;
#pragma unroll
        for (int r = 0; r < 8; ++r) {
            int i = tm * 16 + r + 8 * hi;
            size_t mi = ((size_t)b * NN + i) * NN + colj;
            float val = acc[t4][r] * inv_scale + dis[mi];
            if (mask[mi] == 0) val = -1e9f;
            scores[(sbase + i) * NN + colj] = val;
        }
    }
}

// merged[b,i,j] = sum_h scores[b,h,i,j]  (pre-softmax)
__global__ void merged_sum_kernel(const float* __restrict__ scores,
                                  float* __restrict__ merged) {
    size_t idx = (size_t)blockIdx.x * blockDim.x + threadIdx.x;
    if (idx >= (size_t)BB * NN * NN) return;
    size_t b = idx / ((size_t)NN * NN);
    size_t rem = idx - b * (size_t)NN * NN;
    float s = 0.0f;
#pragma unroll
    for (int h = 0; h < HH; ++h)
        s += scores[((size_t)(b * HH + h)) * NN * NN + rem];
    merged[idx] = s;
}

// In-place row softmax over width NN. Block = 256 threads, 4 elems/thread.
__global__ void softmax_kernel(float* __restrict__ data) {
    size_t row = blockIdx.x;
    float* p = data + row * (size_t)NN;
    int t = threadIdx.x;
    float v[4];
    float m = -3.4e38f;
#pragma unroll
    for (int i = 0; i < 4; ++i) { v[i] = p[t + i * 256]; m = fmaxf(m, v[i]); }
    __shared__ float red[256];
    red[t] = m; __syncthreads();
    for (int off = 128; off > 0; off >>= 1) {
        if (t < off) red[t] = fmaxf(red[t], red[t + off]);
        __syncthreads();
    }
    m = red[0]; __syncthreads();
    float s = 0.0f;
#pragma unroll
    for (int i = 0; i < 4; ++i) { v[i] = __expf(v[i] - m); s += v[i]; }
    red[t] = s; __syncthreads();
    for (int off = 128; off > 0; off >>= 1) {
        if (t < off) red[t] += red[t + off];
        __syncthreads();
    }
    float inv = 1.0f / red[0];
#pragma unroll
    for (int i = 0; i < 4; ++i) p[t + i * 256] = v[i] * inv;
}

// ---------------------------------------------------------------------------
// attn_v[b,i,h*48+c] = sum_j attn[b,h,i,j] * v[b,j,h*48+c]; K=1024.
// V is stored pair-interleaved by the V projection -> both operands b64.
// One wave computes a 16(i) x 48(c) strip (3 accumulators).
// ---------------------------------------------------------------------------
__global__ void attnv_kernel(const float* __restrict__ attn,
                             const float* __restrict__ vpk,
                             float* __restrict__ out) {
    int tm = blockIdx.x;       // 0..63
    int bh = blockIdx.y;
    int b = bh / HH, h = bh % HH;
    int lane = threadIdx.x;
    int hi = lane >> 4, lo = lane & 15;
    const float* ap = attn + (size_t)(b * HH + h) * NN * NN
                           + (size_t)(tm * 16 + lo) * NN;
    const float* vb = vpk + (size_t)b * NN * DD;   // packed pairs within batch b
    v8f acc[3] = {{}, {}, {}};
    for (int k0 = 0; k0 < NN; k0 += 4) {
        v2f a = *(const v2f*)(ap + k0 + 2 * hi);
        const float* wp = vb + ((size_t)((k0 >> 1) + hi) * DD + h * HDC + lo) * 2;
        v2f b0 = *(const v2f*)(wp);
        v2f b1 = *(const v2f*)(wp + 32);
        v2f b2 = *(const v2f*)(wp + 64);
        acc[0] = wmma_f32(a, b0, acc[0]);
        acc[1] = wmma_f32(a, b1, acc[1]);
        acc[2] = wmma_f32(a, b2, acc[2]);
    }
#pragma unroll
    for (int t4 = 0; t4 < 3; ++t4) {
#pragma unroll
        for (int r = 0; r < 8; ++r) {
            int i = tm * 16 + r + 8 * hi;
            out[((size_t)b * NN + i) * DD + h * HDC + 16 * t4 + lo] = acc[t4][r];
        }
    }
}

// v3d[row,c] = h[row,:] . W3[:,c] + b3[c]   (tiny: 2048*3 dots of 384)
__global__ void v3d_kernel(const float* __restrict__ h,
                           const float* __restrict__ W3,
                           const float* __restrict__ b3,
                           float* __restrict__ v3d) {
    int idx = blockIdx.x * blockDim.x + threadIdx.x;
    if (idx >= BB * NN * 3) return;
    int row = idx / 3, cc = idx - row * 3;
    const float* hr = h + (size_t)row * DD;
    float acc = b3[cc];
    for (int kk = 0; kk < DD; ++kk) acc += hr[kk] * W3[kk * 3 + cc];
    v3d[idx] = acc;
}

// frame[b,i,:] = (1/N) sum_j merged[b,i,j] * cross(ad[b,i,j,:], colsum(O)*v3d[b,j,:])
__global__ void frame_kernel(const float* __restrict__ merged,
                             const float* __restrict__ ad,
                             const float* __restrict__ orient,
                             const float* __restrict__ v3d,
                             float* __restrict__ frame) {
    int bi = blockIdx.x;
    int b = bi / NN;
    int t = threadIdx.x;
    __shared__ float sv[NN * 3];           // stage v3d[b,:,:] in LDS (12 KB)
    for (int e = t; e < NN * 3; e += blockDim.x)
        sv[e] = v3d[(size_t)b * NN * 3 + e];
    __syncthreads();
    float a0 = 0.0f, a1 = 0.0f, a2 = 0.0f;
    size_t base = (size_t)bi * NN;
    for (int j = t; j < NN; j += blockDim.x) {
        const float* O = orient + (base + j) * 9;
        const float* A = ad + (base + j) * 3;
        __builtin_prefetch(O + 9 * 256, 0, 1);     // global_prefetch_b8
        float w = merged[base + j];
        float s0 = O[0] + O[3] + O[6];             // column sums over m
        float s1 = O[1] + O[4] + O[7];
        float s2 = O[2] + O[5] + O[8];
        float m0 = s0 * sv[j * 3 + 0];
        float m1 = s1 * sv[j * 3 + 1];
        float m2 = s2 * sv[j * 3 + 2];
        float c0 = A[1] * m2 - A[2] * m1;          // cross(A, M)
        float c1 = A[2] * m0 - A[0] * m2;
        float c2 = A[0] * m1 - A[1] * m0;
        a0 += w * c0; a1 += w * c1; a2 += w * c2;
    }
    __shared__ float r0[256], r1[256], r2[256];
    r0[t] = a0; r1[t] = a1; r2[t] = a2; __syncthreads();
    for (int off = 128; off > 0; off >>= 1) {
        if (t < off) { r0[t] += r0[t + off]; r1[t] += r1[t + off]; r2[t] += r2[t + off]; }
        __syncthreads();
    }
    if (t == 0) {
        frame[(size_t)bi * 3 + 0] = r0[0] * (1.0f / NN);
        frame[(size_t)bi * 3 + 1] = r1[0] * (1.0f / NN);
        frame[(size_t)bi * 3 + 2] = r2[0] * (1.0f / NN);
    }
}

// Pack concat(attn_v, frame) into [row, 388] with zero pad at col 387.
__global__ void pack_fc_kernel(const float* __restrict__ attn_v,
                               const float* __restrict__ frame,
                               float* __restrict__ fcin) {
    int idx = blockIdx.x * blockDim.x + threadIdx.x;
    if (idx >= BB * NN * FCK_PAD) return;
    int row = idx / FCK_PAD, c = idx - row * FCK_PAD;
    float val;
    if (c < DD)            val = attn_v[(size_t)row * DD + c];
    else if (c < DD + 3)   val = frame[(size_t)row * 3 + (c - DD)];
    else                   val = 0.0f;
    fcin[idx] = val;
}

// ---------------------------------------------------------------------------
extern "C" void kernel_launch(void* const* d_in, const int* in_sizes, int n_in,
                              void* d_out, int out_size, void* d_ws, size_t ws_size,
                              hipStream_t stream) {
    const float* x_rigid  = (const float*)d_in[0];
    const float* ad       = (const float*)d_in[1];
    const float* orient   = (const float*)d_in[2];
    const float* distance = (const float*)d_in[3];
    const int*   amask    = (const int*)d_in[4];
    const float* Wq   = (const float*)d_in[5];   const float* bq   = (const float*)d_in[6];
    const float* Wk   = (const float*)d_in[7];   const float* bk   = (const float*)d_in[8];
    const float* Wv   = (const float*)d_in[9];   const float* bv   = (const float*)d_in[10];
    const float* W3   = (const float*)d_in[11];  const float* b3   = (const float*)d_in[12];
    const float* Wmlp = (const float*)d_in[13];  const float* bmlp = (const float*)d_in[14];
    const float* Wfc  = (const float*)d_in[15];  const float* bfc  = (const float*)d_in[16];
    const float* Wff1 = (const float*)d_in[17];  const float* bff1 = (const float*)d_in[18];
    const float* Wff2 = (const float*)d_in[19];  const float* bff2 = (const float*)d_in[20];
    const float* ln1g = (const float*)d_in[21];  const float* ln1b = (const float*)d_in[22];
    const float* ln2g = (const float*)d_in[23];  const float* ln2b = (const float*)d_in[24];

    float* ws = (float*)d_ws;
    size_t off = 0;
    auto carve = [&](size_t n) { float* p = ws + off; off += n; return p; };
    const size_t ROWS = (size_t)BB * NN;            // 2048
    float* xbuf   = carve(ROWS * DD);
    float* hbuf   = carve(ROWS * DD);
    float* qbuf   = carve(ROWS * DD);
    float* kbuf   = carve(ROWS * DD);
    float* vbuf   = carve(ROWS * DD);               // pair-interleaved V
    float* scores = carve((size_t)BB * HH * NN * NN);
    float* merged = carve((size_t)BB * NN * NN);
    float* disb   = carve((size_t)BB * NN * NN);
    float* avbuf  = carve(ROWS * DD);
    float* v3db   = carve(ROWS * 3);
    float* frameb = carve(ROWS * 3);
    float* fcin   = carve(ROWS * FCK_PAD);
    float* ffb    = carve(ROWS * DFFC);
    // packed weights (pair-interleaved, all layers)
    float* WqP   = carve((size_t)LLC * DD * DD);
    float* WkP   = carve((size_t)LLC * DD * DD);
    float* WvP   = carve((size_t)LLC * DD * DD);
    float* Wff1P = carve((size_t)LLC * DD * DFFC);
    float* Wff2P = carve((size_t)LLC * DFFC * DD);
    float* WfcP  = carve((size_t)LLC * FCK_PAD * DD);

    // Pack all weights once (cheap; part of the captured graph, deterministic).
    for (int l = 0; l < LLC; ++l) {
        int gdd = (DD * DD + 255) / 256;
        pack_pairs_kernel<<<gdd, 256, 0, stream>>>(Wq + (size_t)l * DD * DD,
                                                   WqP + (size_t)l * DD * DD, DD, DD, DD);
        pack_pairs_kernel<<<gdd, 256, 0, stream>>>(Wk + (size_t)l * DD * DD,
                                                   WkP + (size_t)l * DD * DD, DD, DD, DD);
        pack_pairs_kernel<<<gdd, 256, 0, stream>>>(Wv + (size_t)l * DD * DD,
                                                   WvP + (size_t)l * DD * DD, DD, DD, DD);
        pack_pairs_kernel<<<(DD * DFFC + 255) / 256, 256, 0, stream>>>(
            Wff1 + (size_t)l * DD * DFFC, Wff1P + (size_t)l * DD * DFFC, DD, DD, DFFC);
        pack_pairs_kernel<<<(DFFC * DD + 255) / 256, 256, 0, stream>>>(
            Wff2 + (size_t)l * DFFC * DD, Wff2P + (size_t)l * DFFC * DD, DFFC, DFFC, DD);
        pack_pairs_kernel<<<(FCK_PAD * DD + 255) / 256, 256, 0, stream>>>(
            Wfc + (size_t)l * FCK * DD, WfcP + (size_t)l * FCK_PAD * DD, FCK, FCK_PAD, DD);
    }

    hipMemcpyAsync(xbuf, x_rigid, ROWS * DD * sizeof(float),
                   hipMemcpyDeviceToDevice, stream);

    const dim3 gD (DD / 64,   ROWS / 16);     // 6 x 128 strips
    const dim3 gFF(DFFC / 64, ROWS / 16);     // 8 x 128 strips
    const dim3 gS (NN / 64, NN / 16, BB * HH);
    const dim3 gAV(NN / 16, BB * HH);

    for (int l = 0; l < LLC; ++l) {
        // h = LN1(x)
        ln_kernel<<<ROWS, 128, 0, stream>>>(xbuf, ln1g + l * DD, ln1b + l * DD, hbuf);
        // q,k,v = h@W + b   (v stored pair-interleaved for attn@V)
        gemm4_kernel<<<gD, 32, 0, stream>>>(hbuf, DD, WqP + (size_t)l * DD * DD, DD,
                                            bq + l * DD, nullptr, qbuf, DD, DD, 0);
        gemm4_kernel<<<gD, 32, 0, stream>>>(hbuf, DD, WkP + (size_t)l * DD * DD, DD,
                                            bk + l * DD, nullptr, kbuf, DD, DD, 0);
        gemm4_kernel<<<gD, 32, 0, stream>>>(hbuf, DD, WvP + (size_t)l * DD * DD, DD,
                                            bv + l * DD, nullptr, vbuf, DD, DD, 4);
        // RBF distance bias
        dis_kernel<<<((size_t)BB * NN * NN + 255) / 256, 256, 0, stream>>>(
            distance, Wmlp + l * RBFC, bmlp + l, disb);
        // scores = qk^T/scale + dis, masked
        scores_kernel<<<gS, 32, 0, stream>>>(qbuf, kbuf, disb, amask, scores);
        // merged pre-softmax head sum
        merged_sum_kernel<<<((size_t)BB * NN * NN + 255) / 256, 256, 0, stream>>>(scores, merged);
        // softmax per head-row, then merged rows
        softmax_kernel<<<(size_t)BB * HH * NN, 256, 0, stream>>>(scores);
        softmax_kernel<<<ROWS, 256, 0, stream>>>(merged);
        // attn @ v
        attnv_kernel<<<gAV, 32, 0, stream>>>(scores, vbuf, avbuf);
        // geometric frame update
        v3d_kernel<<<(BB * NN * 3 + 255) / 256, 256, 0, stream>>>(
            hbuf, W3 + (size_t)l * DD * 3, b3 + l * 3, v3db);
        frame_kernel<<<ROWS, 256, 0, stream>>>(merged, ad, orient, v3db, frameb);
        // x = x + concat(attn_v, frame) @ Wfc + bfc
        pack_fc_kernel<<<(BB * NN * FCK_PAD + 255) / 256, 256, 0, stream>>>(avbuf, frameb, fcin);
        gemm4_kernel<<<gD, 32, 0, stream>>>(fcin, FCK_PAD, WfcP + (size_t)l * FCK_PAD * DD, DD,
                                            bfc + l * DD, xbuf, xbuf, DD, FCK_PAD, 2);
        // FFN with residual
        ln_kernel<<<ROWS, 128, 0, stream>>>(xbuf, ln2g + l * DD, ln2b + l * DD, hbuf);
        gemm4_kernel<<<gFF, 32, 0, stream>>>(hbuf, DD, Wff1P + (size_t)l * DD * DFFC, DFFC,
                                             bff1 + l * DFFC, nullptr, ffb, DFFC, DD, 1);
        gemm4_kernel<<<gD, 32, 0, stream>>>(ffb, DFFC, Wff2P + (size_t)l * DFFC * DD, DD,
                                            bff2 + l * DD, xbuf, xbuf, DD, DFFC, 2);
    }

    hipMemcpyAsync(d_out, xbuf, ROWS * DD * sizeof(float),
                   hipMemcpyDeviceToDevice, stream);
}